// HNHNModel_48584670052999
// MI455X (gfx1250) — compile-verified
//
#include <hip/hip_runtime.h>
#include <cstddef>

typedef __attribute__((ext_vector_type(2))) float v2f;
typedef __attribute__((ext_vector_type(8))) float v8f;

#define N_NODES 200000
#define N_EDGES 400000
#define HID 128
#define XS_STRIDE 132   // 128 + 4 pad: A-frag LDS reads hit 32 distinct banks

// ---------------------------------------------------------------------------
// Normalization: degrees + CSR counts (per-nnz atomics, cheap: 1.8M elements)
// ---------------------------------------------------------------------------
__global__ void deg_count_kernel(const int* __restrict__ rows, const int* __restrict__ cols,
                                 const float* __restrict__ vals,
                                 float* __restrict__ deg_v, float* __restrict__ deg_e,
                                 int* __restrict__ row_cnt, int* __restrict__ col_cnt, int nnz) {
  int i = blockIdx.x * blockDim.x + threadIdx.x;
  if (i >= nnz) return;
  int r = rows[i], c = cols[i];
  float v = vals[i];
  atomicAdd(&deg_v[r], v);
  atomicAdd(&deg_e[c], v);
  atomicAdd(&row_cnt[r], 1);
  atomicAdd(&col_cnt[c], 1);
}

// deg_v -> deg_v^BETA (=rsqrt), deg_e -> deg_e^ALPHA (= rsqrt(d)/d), in place
__global__ void card_kernel(float* __restrict__ deg_v, float* __restrict__ deg_e) {
  int i = blockIdx.x * blockDim.x + threadIdx.x;
  if (i < N_NODES) {
    float d = deg_v[i];
    deg_v[i] = rsqrtf(d);              // d^-0.5
  }
  if (i < N_EDGES) {
    float d = deg_e[i];
    deg_e[i] = rsqrtf(d) / d;          // d^-1.5
  }
}

__global__ void denom_kernel(const int* __restrict__ rows, const int* __restrict__ cols,
                             const float* __restrict__ vals,
                             const float* __restrict__ n_card, const float* __restrict__ e_card,
                             float* __restrict__ denom_v, float* __restrict__ denom_e, int nnz) {
  int i = blockIdx.x * blockDim.x + threadIdx.x;
  if (i >= nnz) return;
  int r = rows[i], c = cols[i];
  float v = vals[i];
  atomicAdd(&denom_v[r], v * e_card[c]);
  atomicAdd(&denom_e[c], v * n_card[r]);
}

// ---------------------------------------------------------------------------
// Single-block chunked exclusive scan (8 elems/thread, 1024 threads per chunk)
// Writes exclusive prefix to ptr[0..n] and a working copy to ofs[0..n-1].
// ---------------------------------------------------------------------------
__global__ void scan_kernel(const int* __restrict__ cnt, int* __restrict__ ptr,
                            int* __restrict__ ofs, int n) {
  __shared__ int sm[1024];
  __shared__ int carry_s;
  if (threadIdx.x == 0) carry_s = 0;
  __syncthreads();
  const int CHUNK = 1024 * 8;
  for (int base = 0; base < n; base += CHUNK) {
    int loc[8];
    int tsum = 0;
    int i0 = base + (int)threadIdx.x * 8;
    #pragma unroll
    for (int j = 0; j < 8; ++j) {
      int i = i0 + j;
      int v = (i < n) ? cnt[i] : 0;
      loc[j] = tsum;                     // thread-local exclusive
      tsum += v;
    }
    sm[threadIdx.x] = tsum;
    __syncthreads();
    for (int off = 1; off < 1024; off <<= 1) {
      int t = (threadIdx.x >= (unsigned)off) ? sm[threadIdx.x - off] : 0;
      __syncthreads();
      sm[threadIdx.x] += t;
      __syncthreads();
    }
    int texcl = sm[threadIdx.x] - tsum;  // exclusive prefix of this thread in chunk
    int carry = carry_s;
    #pragma unroll
    for (int j = 0; j < 8; ++j) {
      int i = i0 + j;
      if (i < n) {
        int e = carry + texcl + loc[j];
        ptr[i] = e;
        ofs[i] = e;
      }
    }
    __syncthreads();
    if (threadIdx.x == 1023) carry_s += sm[1023];
    __syncthreads();
  }
  if (threadIdx.x == 0) ptr[n] = carry_s;
}

// ---------------------------------------------------------------------------
// CSR placement with normalized values (both directions at once)
// ---------------------------------------------------------------------------
__global__ void place_kernel(const int* __restrict__ rows, const int* __restrict__ cols,
                             const float* __restrict__ vals,
                             const float* __restrict__ n_card, const float* __restrict__ e_card,
                             const float* __restrict__ denom_v, const float* __restrict__ denom_e,
                             int* __restrict__ col_ofs, int* __restrict__ row_ofs,
                             int* __restrict__ col_rows, float* __restrict__ col_vals,
                             int* __restrict__ row_cols, float* __restrict__ row_vals, int nnz) {
  int i = blockIdx.x * blockDim.x + threadIdx.x;
  if (i >= nnz) return;
  int r = rows[i], c = cols[i];
  float v = vals[i];
  float vfe = v * e_card[c] / denom_v[r];   // node <- edge weight
  float efv = v * n_card[r] / denom_e[c];   // edge <- node weight
  int p = atomicAdd(&col_ofs[c], 1);
  col_rows[p] = r;
  col_vals[p] = efv;
  int q = atomicAdd(&row_ofs[r], 1);
  row_cols[q] = c;
  row_vals[q] = vfe;
}

// ---------------------------------------------------------------------------
// Layer-1 input GEMM: [N_NODES x 14] @ [14 x 128]  (tiny K, plain VALU)
// ---------------------------------------------------------------------------
__global__ void gemm_in14(const float* __restrict__ X, const float* __restrict__ W,
                          float* __restrict__ Y) {
  __shared__ float xr[14];
  int row = blockIdx.x;
  if (threadIdx.x < 14) xr[threadIdx.x] = X[row * 14 + threadIdx.x];
  __syncthreads();
  float acc = 0.f;
  #pragma unroll
  for (int k = 0; k < 14; ++k) acc += xr[k] * W[k * HID + threadIdx.x];
  Y[(size_t)row * HID + threadIdx.x] = acc;
}

// ---------------------------------------------------------------------------
// WMMA fp32 GEMM: Y[R x 128] = X[R x 128] @ W[128 x 128]
// Block = 256 threads = 8 waves.  The 16x128 X stripe is staged in LDS once
// (coalesced, 8KB) so the 8 waves don't each re-read it from HBM (X is 205MB,
// bigger than the 192MB L2 -> 8x redundancy would leak to HBM).
// Wave w owns output tile (16 rows x cols 16w..16w+15); K=128 -> 32 WMMAs.
// ---------------------------------------------------------------------------
__global__ void wmma_gemm_128(const float* __restrict__ X, const float* __restrict__ W,
                              float* __restrict__ Y) {
  __shared__ float xs[16 * XS_STRIDE];

  const int row0 = blockIdx.x * 16;
  // cooperative coalesced stage of X[row0..row0+15][0..127] into LDS
  for (int idx = threadIdx.x; idx < 16 * HID; idx += 256) {
    int r = idx >> 7;
    int k = idx & (HID - 1);
    xs[r * XS_STRIDE + k] = X[(size_t)(row0 + r) * HID + k];
  }
  __syncthreads();

  const int lane  = threadIdx.x & 31;
  const int wave  = threadIdx.x >> 5;          // 0..7 -> column tile
  const int col0  = wave * 16;
  const int l15   = lane & 15;
  const int hi    = lane >> 4;                 // 0: K pair {0,1}; 1: K pair {2,3}
  const int nc    = col0 + l15;                // B: N = lane
  const int khalf = hi * 2;

  const float* __restrict__ xrow = xs + l15 * XS_STRIDE;  // A: M = lane
  v8f c = {};
  #pragma unroll
  for (int k0 = 0; k0 < HID; k0 += 4) {
    int kk = k0 + khalf;
    v2f a = *(const v2f*)(xrow + kk);          // (M, K=kk), (M, K=kk+1) from LDS
    v2f b;
    b[0] = W[kk * HID + nc];                   // (K=kk,   N=nc)
    b[1] = W[(kk + 1) * HID + nc];             // (K=kk+1, N=nc)
    c = __builtin_amdgcn_wmma_f32_16x16x4_f32(false, a, false, b, (short)0, c,
                                              false, false);
  }
  // D layout: VGPR r, lanes 0-15 -> M=r; lanes 16-31 -> M=r+8; N = lane&15
  const int mbase = row0 + 8 * hi;
  #pragma unroll
  for (int r = 0; r < 8; ++r) {
    Y[(size_t)(mbase + r) * HID + nc] = c[r];
  }
}

// ---------------------------------------------------------------------------
// Gather SpMM + bias + sigmoid.  One block (128 threads) per output row;
// thread = channel -> each nnz is one coalesced 512B row read.
// ---------------------------------------------------------------------------
__global__ void spmm_sigmoid(const int* __restrict__ ptr, const int* __restrict__ idx,
                             const float* __restrict__ w, const float* __restrict__ Min,
                             const float* __restrict__ bias, float* __restrict__ Out) {
  int o  = blockIdx.x;
  int ch = threadIdx.x;
  int s = ptr[o], e = ptr[o + 1];
  float acc = 0.f;
  for (int j = s; j < e; ++j) {
    acc += w[j] * Min[(size_t)idx[j] * HID + ch];
  }
  float z = acc + bias[ch];
  Out[(size_t)o * HID + ch] = 1.f / (1.f + __expf(-z));
}

// ---------------------------------------------------------------------------
// Column max-pool over N_NODES, then pooled @ lin_w + lin_b
// ---------------------------------------------------------------------------
__global__ void maxpool_kernel(const float* __restrict__ X, float* __restrict__ pooled) {
  int ch = blockIdx.x;               // 128 blocks
  float m = -3.402823466e38f;
  for (int i = threadIdx.x; i < N_NODES; i += blockDim.x)
    m = fmaxf(m, X[(size_t)i * HID + ch]);
  __shared__ float sm[256];
  sm[threadIdx.x] = m;
  __syncthreads();
  for (int off = 128; off > 0; off >>= 1) {
    if (threadIdx.x < (unsigned)off) sm[threadIdx.x] = fmaxf(sm[threadIdx.x], sm[threadIdx.x + off]);
    __syncthreads();
  }
  if (threadIdx.x == 0) pooled[ch] = sm[0];
}

__global__ void final_linear(const float* __restrict__ pooled, const float* __restrict__ lw,
                             const float* __restrict__ lb, float* __restrict__ out) {
  __shared__ float sm[HID];
  sm[threadIdx.x] = pooled[threadIdx.x] * lw[threadIdx.x];
  __syncthreads();
  for (int off = 64; off > 0; off >>= 1) {
    if (threadIdx.x < (unsigned)off) sm[threadIdx.x] += sm[threadIdx.x + off];
    __syncthreads();
  }
  if (threadIdx.x == 0) out[0] = sm[0] + lb[0];
}

// ---------------------------------------------------------------------------
extern "C" void kernel_launch(void* const* d_in, const int* in_sizes, int n_in,
                              void* d_out, int out_size, void* d_ws, size_t ws_size,
                              hipStream_t stream) {
  (void)n_in; (void)out_size; (void)ws_size;

  const float* x0_in   = (const float*)d_in[0];
  const int*   rows    = (const int*)  d_in[1];
  const int*   cols    = (const int*)  d_in[2];
  const float* vals    = (const float*)d_in[3];
  const float* w0_l1   = (const float*)d_in[4];
  const float* w1_l1   = (const float*)d_in[5];
  const float* b1_l1   = (const float*)d_in[6];
  const float* b0_l1   = (const float*)d_in[7];
  const float* w0_l2   = (const float*)d_in[8];
  const float* w1_l2   = (const float*)d_in[9];
  const float* b1_l2   = (const float*)d_in[10];
  const float* b0_l2   = (const float*)d_in[11];
  const float* lin_w   = (const float*)d_in[12];
  const float* lin_b   = (const float*)d_in[13];
  float* out = (float*)d_out;

  const int nnz = in_sizes[1];

  // ---- workspace carve-up (256B aligned slices) ----
  char* cur = (char*)d_ws;
  auto carve = [&](size_t bytes) -> void* {
    void* p = (void*)cur;
    size_t b = (bytes + 255) & ~(size_t)255;
    cur += b;
    return p;
  };
  float* deg_v    = (float*)carve(sizeof(float) * N_NODES);   // -> n_card in place
  float* deg_e    = (float*)carve(sizeof(float) * N_EDGES);   // -> e_card in place
  float* denom_v  = (float*)carve(sizeof(float) * N_NODES);
  float* denom_e  = (float*)carve(sizeof(float) * N_EDGES);
  int*   row_cnt  = (int*)  carve(sizeof(int) * N_NODES);
  int*   col_cnt  = (int*)  carve(sizeof(int) * N_EDGES);
  int*   row_ptr  = (int*)  carve(sizeof(int) * (N_NODES + 1));
  int*   col_ptr  = (int*)  carve(sizeof(int) * (N_EDGES + 1));
  int*   row_ofs  = (int*)  carve(sizeof(int) * N_NODES);
  int*   col_ofs  = (int*)  carve(sizeof(int) * N_EDGES);
  int*   col_rows = (int*)  carve(sizeof(int) * nnz);
  float* col_vals = (float*)carve(sizeof(float) * nnz);
  int*   row_cols = (int*)  carve(sizeof(int) * nnz);
  float* row_vals = (float*)carve(sizeof(float) * nnz);
  float* m_node   = (float*)carve(sizeof(float) * (size_t)N_NODES * HID);
  float* x0_node  = (float*)carve(sizeof(float) * (size_t)N_NODES * HID);
  float* x1_edge  = (float*)carve(sizeof(float) * (size_t)N_EDGES * HID);
  float* m2_edge  = (float*)carve(sizeof(float) * (size_t)N_EDGES * HID);
  float* pooled   = (float*)carve(sizeof(float) * HID);

  // ---- zero the accumulators (graph-capture-safe async memsets) ----
  hipMemsetAsync(deg_v,   0, sizeof(float) * N_NODES, stream);
  hipMemsetAsync(deg_e,   0, sizeof(float) * N_EDGES, stream);
  hipMemsetAsync(denom_v, 0, sizeof(float) * N_NODES, stream);
  hipMemsetAsync(denom_e, 0, sizeof(float) * N_EDGES, stream);
  hipMemsetAsync(row_cnt, 0, sizeof(int) * N_NODES, stream);
  hipMemsetAsync(col_cnt, 0, sizeof(int) * N_EDGES, stream);

  const int TB = 256;
  const int nnz_blocks = (nnz + TB - 1) / TB;

  // ---- normalization + CSR build ----
  deg_count_kernel<<<nnz_blocks, TB, 0, stream>>>(rows, cols, vals, deg_v, deg_e,
                                                  row_cnt, col_cnt, nnz);
  card_kernel<<<(N_EDGES + TB - 1) / TB, TB, 0, stream>>>(deg_v, deg_e);
  denom_kernel<<<nnz_blocks, TB, 0, stream>>>(rows, cols, vals, deg_v, deg_e,
                                              denom_v, denom_e, nnz);
  scan_kernel<<<1, 1024, 0, stream>>>(col_cnt, col_ptr, col_ofs, N_EDGES);
  scan_kernel<<<1, 1024, 0, stream>>>(row_cnt, row_ptr, row_ofs, N_NODES);
  place_kernel<<<nnz_blocks, TB, 0, stream>>>(rows, cols, vals, deg_v, deg_e,
                                              denom_v, denom_e, col_ofs, row_ofs,
                                              col_rows, col_vals, row_cols, row_vals, nnz);

  // ---- layer 1 ----
  gemm_in14<<<N_NODES, HID, 0, stream>>>(x0_in, w0_l1, m_node);
  spmm_sigmoid<<<N_EDGES, HID, 0, stream>>>(col_ptr, col_rows, col_vals, m_node, b1_l1, x1_edge);
  wmma_gemm_128<<<N_EDGES / 16, 256, 0, stream>>>(x1_edge, w1_l1, m2_edge);
  spmm_sigmoid<<<N_NODES, HID, 0, stream>>>(row_ptr, row_cols, row_vals, m2_edge, b0_l1, x0_node);

  // ---- layer 2 ----
  wmma_gemm_128<<<N_NODES / 16, 256, 0, stream>>>(x0_node, w0_l2, m_node);
  spmm_sigmoid<<<N_EDGES, HID, 0, stream>>>(col_ptr, col_rows, col_vals, m_node, b1_l2, x1_edge);
  wmma_gemm_128<<<N_EDGES / 16, 256, 0, stream>>>(x1_edge, w1_l2, m2_edge);
  spmm_sigmoid<<<N_NODES, HID, 0, stream>>>(row_ptr, row_cols, row_vals, m2_edge, b0_l2, x0_node);

  // ---- pool + head ----
  maxpool_kernel<<<HID, 256, 0, stream>>>(x0_node, pooled);
  final_linear<<<1, HID, 0, stream>>>(pooled, lin_w, lin_b, out);
}